// GGNN_43215960933044
// MI455X (gfx1250) — compile-verified
//
#include <hip/hip_runtime.h>
#include <math.h>

// ---------------------------------------------------------------------------
// Fused GGNN (B=8192, N=82, H=10, OUT=5, T=3) for gfx1250.
//  * 1 workgroup = 256 threads = 8 wave32 = 8 batch elements.
//  * Node state (transposed), adjacency and weights live in LDS for the whole
//    T-step recurrence (batches independent -> no grid sync).
//  * Adjacency aggregation uses V_WMMA_F32_16X16X4_F32 with fully zero-padded
//    LDS operands: the inner loop is 2x ds_load_b64 + v_wmma, no predication.
//  * Gate GEMVs (20->10, 10->10) are VALU FMAs (too skinny for 16x16 tiles).
// ---------------------------------------------------------------------------

typedef __attribute__((ext_vector_type(2))) float v2f;
typedef __attribute__((ext_vector_type(8))) float v8f;

constexpr int kB = 8192, kN = 82, kH = 10, kOUT = 5, kT = 3;
constexpr int NB = 8;               // batches per workgroup (== #waves)
constexpr int ADJ_LD = 84;          // K padded to 21*4
constexpr int ADJ_ROWS = 96;        // M padded to 6*16 (rows 82..95 zero)
constexpr int NT_ROWS = 16;         // h padded to 16 (rows 10..15 zero)
constexpr int NT_LD = 84;           // j padded (cols 82..83 zero)
constexpr int AV_LD = 32;           // a_in cols 0..15, a_out cols 16..31
constexpr int AV_ROWS = 96;         // i padded to 6*16

// LDS float-offset map
constexpr int OFF_ADJ_IN  = 0;                                   // 96*84 = 8064
constexpr int OFF_ADJ_OUT = OFF_ADJ_IN + ADJ_ROWS * ADJ_LD;      // 8064
constexpr int OFF_NODES   = OFF_ADJ_OUT + ADJ_ROWS * ADJ_LD;     // 8*16*84 = 10752
constexpr int OFF_AV      = OFF_NODES + NB * NT_ROWS * NT_LD;    // 8*96*32 = 24576
constexpr int OFF_W       = OFF_AV + NB * AV_ROWS * AV_LD;
constexpr int W_TOTAL     = 960;
constexpr int SMEM_FLOATS = OFF_W + W_TOTAL;                     // 52416 fl ~ 205 KB

// weight sub-offsets inside sW
constexpr int W3W = 0,   B3W = 200, W3U = 210, B3U = 310;
constexpr int W4W = 320, B4W = 520, W5W = 530, B5W = 730;
constexpr int W5U = 740, B5U = 840, WOUT = 850, BOUT = 950;

__global__ __launch_bounds__(256) void ggnn_fused_wmma(
    const float* __restrict__ x,
    const float* __restrict__ in_adj, const float* __restrict__ out_adj,
    const float* __restrict__ w3w, const float* __restrict__ b3w,
    const float* __restrict__ w3u, const float* __restrict__ b3u,
    const float* __restrict__ w4w, const float* __restrict__ b4w,
    const float* __restrict__ w5w, const float* __restrict__ b5w,
    const float* __restrict__ w5u, const float* __restrict__ b5u,
    const float* __restrict__ wout, const float* __restrict__ bout,
    float* __restrict__ out, float* __restrict__ fn_out)
{
  extern __shared__ float smem[];
  float* sAdjIn  = smem + OFF_ADJ_IN;
  float* sAdjOut = smem + OFF_ADJ_OUT;
  float* sNodes  = smem + OFF_NODES;   // per batch: transposed [16][84]
  float* sAv     = smem + OFF_AV;      // per batch: [96][32]
  float* sW      = smem + OFF_W;

  const int tid = threadIdx.x;
  const long bbase = (long)blockIdx.x * NB;

  // ---- stage adjacency into LDS (rows/cols zero-padded) ----
  for (int idx = tid; idx < ADJ_ROWS * ADJ_LD; idx += 256) {
    const int r = idx / ADJ_LD, c = idx - r * ADJ_LD;
    float vi = 0.f, vo = 0.f;
    if (r < kN && c < kN) { vi = in_adj[r * kN + c]; vo = out_adj[r * kN + c]; }
    sAdjIn[idx] = vi; sAdjOut[idx] = vo;
  }
  // ---- stage weights (all tiny, one conditional copy each) ----
  if (tid < 200) sW[W3W + tid]  = w3w[tid];
  if (tid < 10)  sW[B3W + tid]  = b3w[tid];
  if (tid < 100) sW[W3U + tid]  = w3u[tid];
  if (tid < 10)  sW[B3U + tid]  = b3u[tid];
  if (tid < 200) sW[W4W + tid]  = w4w[tid];
  if (tid < 10)  sW[B4W + tid]  = b4w[tid];
  if (tid < 200) sW[W5W + tid]  = w5w[tid];
  if (tid < 10)  sW[B5W + tid]  = b5w[tid];
  if (tid < 100) sW[W5U + tid]  = w5u[tid];
  if (tid < 10)  sW[B5U + tid]  = b5u[tid];
  if (tid < 100) sW[WOUT + tid] = wout[tid];
  if (tid < 5)   sW[BOUT + tid] = bout[tid];

  // ---- stage node state transposed: nT[bl][h][j] = x[b][j][h] ----
  for (int idx = tid; idx < NB * NT_ROWS * NT_LD; idx += 256) {
    const int bl  = idx / (NT_ROWS * NT_LD);
    const int rem = idx - bl * (NT_ROWS * NT_LD);
    const int h = rem / NT_LD, j = rem - h * NT_LD;
    float v = 0.f;
    if (h < kH && j < kN) v = x[((bbase + bl) * kN + j) * kH + h];
    sNodes[idx] = v;
  }
  // warm L2 for the output-phase x reload (emits global_prefetch_b8)
  if (tid * 32 < NB * kN * kH)
    __builtin_prefetch(x + bbase * kN * kH + tid * 32, 0, 1);
  __syncthreads();

  const int wv   = tid >> 5;          // wave == local batch
  const int lane = tid & 31;
  const int lrow = lane & 15;
  const int hi   = lane >> 4;         // selects K-pair {k0,k0+1} vs {k0+2,k0+3}
  float* nT  = sNodes + wv * (NT_ROWS * NT_LD);
  float* avb = sAv    + wv * (AV_ROWS * AV_LD);
  const float* bptr = nT + lrow * NT_LD;           // B-frag row base (col n=lrow)

  for (int t = 0; t < kT; ++t) {
    // ================= neighbor aggregation via f32 WMMA =================
    // a[i,h] = sum_j adj[i,j] * nodes[j,h]; fully padded -> no predication.
    for (int sel = 0; sel < 2; ++sel) {
      const float* A = sel ? sAdjOut : sAdjIn;
      float* dptr = avb + hi * 8 * AV_LD + sel * 16 + lrow;
      for (int mt = 0; mt < 6; ++mt) {
        const float* aptr = A + (mt * 16 + lrow) * ADJ_LD;
        v8f acc = {};
        #pragma unroll
        for (int k0 = 0; k0 < ADJ_LD; k0 += 4) {
          const int kk = k0 + 2 * hi;
          const v2f af = *(const v2f*)(aptr + kk);   // A[m][kk..kk+1]
          const v2f bf = *(const v2f*)(bptr + kk);   // B[kk..kk+1][n] (transposed)
          acc = __builtin_amdgcn_wmma_f32_16x16x4_f32(
              false, af, false, bf, (short)0, acc, false, false);
        }
        // D layout: VGPR e -> (M = mt*16 + e + hi*8, N = lrow); av padded -> no guard
        #pragma unroll
        for (int e = 0; e < 8; ++e)
          dptr[(mt * 16 + e) * AV_LD] = acc[e];
      }
    }
    __syncthreads();

    // ================= GRU-style gate update (VALU) =================
    for (int r = lane; r < kN; r += 32) {
      float avl[2 * kH], f[kH];
      const float* avr = avb + r * AV_LD;
      #pragma unroll
      for (int k = 0; k < kH; ++k) { avl[k] = avr[k]; avl[kH + k] = avr[16 + k]; }
      #pragma unroll
      for (int k = 0; k < kH; ++k) f[k] = nT[k * NT_LD + r];

      float z[kH], rv[kH], ah[kH];
      #pragma unroll
      for (int h = 0; h < kH; ++h) {
        float az = sW[B3W + h] + sW[B3U + h];
        float ar = sW[B4W + h] + sW[B3U + h];   // reference bug: reuses b3u/w3u
        float a5 = sW[B5W + h] + sW[B5U + h];
        #pragma unroll
        for (int k = 0; k < 2 * kH; ++k) {
          az += avl[k] * sW[W3W + h * 2 * kH + k];
          ar += avl[k] * sW[W4W + h * 2 * kH + k];
          a5 += avl[k] * sW[W5W + h * 2 * kH + k];
        }
        #pragma unroll
        for (int k = 0; k < kH; ++k) {
          const float fu = f[k] * sW[W3U + h * kH + k];
          az += fu; ar += fu;
        }
        z[h]  = 1.f / (1.f + __expf(-az));
        rv[h] = 1.f / (1.f + __expf(-ar));
        ah[h] = a5;
      }
      float rf[kH];
      #pragma unroll
      for (int k = 0; k < kH; ++k) rf[k] = rv[k] * f[k];
      #pragma unroll
      for (int h = 0; h < kH; ++h) {
        float a5 = ah[h];
        #pragma unroll
        for (int k = 0; k < kH; ++k) a5 += rf[k] * sW[W5U + h * kH + k];
        const float hv = tanhf(a5);
        nT[h * NT_LD + r] = (1.f - z[h]) * f[h] + z[h] * hv;
      }
    }
    __syncthreads();
  }

  // ================= output head: tanh([fn, x] @ wout^T + bout) ============
  for (int r = lane; r < kN; r += 32) {
    float f[kH], xr[kH];
    const long row = (bbase + wv) * kN + r;
    #pragma unroll
    for (int k = 0; k < kH; ++k) {
      f[k]  = nT[k * NT_LD + r];
      xr[k] = x[row * kH + k];
    }
    #pragma unroll
    for (int o = 0; o < kOUT; ++o) {
      float a = sW[BOUT + o];
      #pragma unroll
      for (int k = 0; k < kH; ++k)
        a += f[k] * sW[WOUT + o * 2 * kH + k]
           + xr[k] * sW[WOUT + o * 2 * kH + kH + k];
      out[row * kOUT + o] = tanhf(a);
    }
    #pragma unroll
    for (int k = 0; k < kH; ++k) fn_out[row * kH + k] = f[k];
  }
}

extern "C" void kernel_launch(void* const* d_in, const int* in_sizes, int n_in,
                              void* d_out, int out_size, void* d_ws, size_t ws_size,
                              hipStream_t stream) {
  (void)in_sizes; (void)n_in; (void)out_size; (void)d_ws; (void)ws_size;
  const float* x       = (const float*)d_in[0];
  const float* in_adj  = (const float*)d_in[1];
  const float* out_adj = (const float*)d_in[2];
  const float* w3w  = (const float*)d_in[3];
  const float* b3w  = (const float*)d_in[4];
  const float* w3u  = (const float*)d_in[5];
  const float* b3u  = (const float*)d_in[6];
  const float* w4w  = (const float*)d_in[7];
  const float* b4w  = (const float*)d_in[8];
  const float* w5w  = (const float*)d_in[9];
  const float* b5w  = (const float*)d_in[10];
  const float* w5u  = (const float*)d_in[11];
  const float* b5u  = (const float*)d_in[12];
  const float* wout = (const float*)d_in[13];
  const float* bout = (const float*)d_in[14];

  float* out    = (float*)d_out;
  float* fn_out = out + (size_t)kB * kN * kOUT;

  const size_t smem = (size_t)SMEM_FLOATS * sizeof(float);  // ~205 KB dynamic LDS
  dim3 grid(kB / NB), block(256);
  ggnn_fused_wmma<<<grid, block, smem, stream>>>(
      x, in_adj, out_adj, w3w, b3w, w3u, b3u, w4w, b4w,
      w5w, b5w, w5u, b5u, wout, bout, out, fn_out);
}